// SoapLayer_74809740361917
// MI455X (gfx1250) — compile-verified
//
#include <hip/hip_runtime.h>
#include <math.h>

// ---------------- constants / types ----------------
#define NATOM_PER_SYS 512
#define CHUNK 128
#define PITCH 132   // LDS row pitch (floats): 132 mod 64 = 4 -> conflict-free WMMA fetches

typedef __attribute__((ext_vector_type(2))) float v2f;
typedef __attribute__((ext_vector_type(8))) float v8f;

struct SoapConsts {
    float Wn[64];     // W_ORTHO[n][k] / radial_norm[k]
    float ycoef[32];  // real-sph-harm coefficients * 4pi (25 used, padded)
    float psn[8];     // PS_NORM[l] = pi*sqrt(8/(2l+1)) (5 used)
};

// upper-triangle (n,p) index tables for 8x8, row-major triu order
__constant__ unsigned char IU_N[36] = {0,0,0,0,0,0,0,0, 1,1,1,1,1,1,1, 2,2,2,2,2,2,
                                       3,3,3,3,3, 4,4,4,4, 5,5,5, 6,6, 7};
__constant__ unsigned char IU_P[36] = {0,1,2,3,4,5,6,7, 1,2,3,4,5,6,7, 2,3,4,5,6,7,
                                       3,4,5,6,7, 4,5,6,7, 5,6,7, 6,7, 7};
// species pairs: combinations_with_replacement(3,2)
__constant__ unsigned char PR_S[6] = {0,0,0,1,1,2};
__constant__ unsigned char PR_T[6] = {0,1,2,1,2,2};

// ---------------- kernel ----------------
__global__ __launch_bounds__(128)
void soap_wmma_kernel(const float* __restrict__ pos, const int* __restrict__ Z,
                      float* __restrict__ out, SoapConsts cc)
{
    __shared__ float sGT[32][PITCH];  // A: G^T, rows = s*8+n (24 live, 8 zero), cols = j-in-chunk
    __shared__ float sYT[32][PITCH];  // B^T: rows = m (25 live, 7 zero), cols = j-in-chunk
    __shared__ float sC[32][33];      // c accumulator tile (rows s*8+n, cols m)

    const int t    = threadIdx.x;
    const int lane = t & 31;
    const int wave = t >> 5;
    const int mt   = wave >> 1;   // M tile (rows of c)
    const int nt   = wave & 1;    // N tile (cols of c)

    const int atom = blockIdx.x;        // = b*512 + i
    const int bidx = atom >> 9;
    const int jbase = bidx * NATOM_PER_SYS;

    const float pix = pos[atom * 3 + 0];
    const float piy = pos[atom * 3 + 1];
    const float piz = pos[atom * 3 + 2];

    // zero padding rows once (never rewritten)
    #pragma unroll
    for (int r = 24; r < 32; ++r) sGT[r][t] = 0.0f;
    #pragma unroll
    for (int m = 25; m < 32; ++m) sYT[m][t] = 0.0f;

    v8f acc = {};  // 16x16 fp32 c-tile, accumulated over all 512 neighbors

    for (int ch = 0; ch < NATOM_PER_SYS / CHUNK; ++ch) {
        const int gj = jbase + ch * CHUNK + t;
        const float dx = pix - pos[gj * 3 + 0];
        const float dy = piy - pos[gj * 3 + 1];
        const float dz = piz - pos[gj * 3 + 2];
        const float d2 = dx * dx + dy * dy + dz * dz;
        const float dist = (d2 > 0.0f) ? sqrtf(d2) : 0.0f;
        const float inv  = (dist > 0.0f) ? (1.0f / dist) : 0.0f;
        const float x = dx * inv, y = dy * inv, z = dz * inv;

        // radial channels: g[n] = sum_k Wn[n][k] * dr^(k+3)   (dr=0 outside cutoff)
        const float dr = fmaxf(5.0f - dist, 0.0f);
        float g[8];
        #pragma unroll
        for (int n = 0; n < 8; ++n) g[n] = 0.0f;
        float pk = dr * dr * dr;
        #pragma unroll
        for (int k = 0; k < 8; ++k) {
            #pragma unroll
            for (int n = 0; n < 8; ++n) g[n] = fmaf(cc.Wn[n * 8 + k], pk, g[n]);
            pk *= dr;
        }
        const int zj = Z[gj];
        const int sj = (zj == 1) ? 0 : ((zj == 6) ? 1 : 2);
        #pragma unroll
        for (int r = 0; r < 24; ++r)
            sGT[r][t] = ((r >> 3) == sj) ? g[r & 7] : 0.0f;

        // real spherical harmonics (homogeneous poly form) * 4pi
        const float x2 = x * x, y2 = y * y, z2 = z * z;
        const float r2 = x2 + y2 + z2;
        float P[25];
        P[0] = 1.0f; P[1] = y; P[2] = z; P[3] = x;
        P[4] = x * y; P[5] = y * z; P[6] = 3.0f * z2 - r2; P[7] = x * z; P[8] = x2 - y2;
        P[9]  = y * (3.0f * x2 - y2);
        P[10] = x * y * z;
        P[11] = y * (5.0f * z2 - r2);
        P[12] = z * (5.0f * z2 - 3.0f * r2);
        P[13] = x * (5.0f * z2 - r2);
        P[14] = z * (x2 - y2);
        P[15] = x * (x2 - 3.0f * y2);
        P[16] = x * y * (x2 - y2);
        P[17] = y * z * (3.0f * x2 - y2);
        P[18] = x * y * (7.0f * z2 - r2);
        P[19] = y * z * (7.0f * z2 - 3.0f * r2);
        P[20] = 35.0f * z2 * z2 - 30.0f * z2 * r2 + 3.0f * r2 * r2;
        P[21] = x * z * (7.0f * z2 - 3.0f * r2);
        P[22] = (x2 - y2) * (7.0f * z2 - r2);
        P[23] = x * z * (x2 - 3.0f * y2);
        P[24] = x2 * x2 - 6.0f * x2 * y2 + y2 * y2;
        #pragma unroll
        for (int m = 0; m < 25; ++m) sYT[m][t] = cc.ycoef[m] * P[m];

        __syncthreads();

        // c_tile += G^T(16x4) x Y(4x16) over this chunk, K in steps of 4.
        // fp32 WMMA A layout: lanes 0-15 hold K={0,1}, lanes 16-31 hold K={2,3}.
        const int Mr   = mt * 16 + (lane & 15);
        const int Nc   = nt * 16 + (lane & 15);
        const int koff = (lane >> 4) << 1;
        #pragma unroll 4
        for (int kk = 0; kk < CHUNK; kk += 4) {
            v2f a  = *(const v2f*)&sGT[Mr][kk + koff];
            v2f bb = *(const v2f*)&sYT[Nc][kk + koff];
            acc = __builtin_amdgcn_wmma_f32_16x16x4_f32(
                      /*neg_a=*/false, a, /*neg_b=*/false, bb,
                      /*c_mod=*/(short)0, acc, /*reuse_a=*/false, /*reuse_b=*/false);
        }
        __syncthreads();
    }

    // spill c-tile to LDS (C/D layout: VGPR r -> M=r (lanes 0-15) / M=r+8 (lanes 16-31))
    {
        const int col   = nt * 16 + (lane & 15);
        const int rbase = mt * 16 + ((lane < 16) ? 0 : 8);
        #pragma unroll
        for (int r = 0; r < 8; ++r) sC[rbase + r][col] = acc[r];
    }
    __syncthreads();

    // power spectrum: out[pair, l, triu(n,p)] with symmetry/normalization factors
    const float SQ2 = 1.41421356237309515f;
    float* outp = out + (size_t)atom * 1080;
    for (int o = t; o < 1080; o += 128) {
        const int pair = o / 180;
        const int rem  = o - pair * 180;
        const int l    = rem / 36;
        const int iu   = rem - l * 36;
        const int n = IU_N[iu], p = IU_P[iu];
        const int s = PR_S[pair], tt = PR_T[pair];
        const int m0 = l * l, mc = 2 * l + 1;
        const float* c0 = &sC[s * 8 + n][m0];
        const float* c1 = &sC[tt * 8 + p][m0];
        float v = 0.0f;
        for (int m = 0; m < mc; ++m) v = fmaf(c0[m], c1[m], v);
        float fac = cc.psn[l];
        if (s != tt) fac *= SQ2;
        if (n < p)   fac *= SQ2;
        outp[o] = v * fac;
    }
}

// ---------------- host ----------------
extern "C" void kernel_launch(void* const* d_in, const int* in_sizes, int n_in,
                              void* d_out, int out_size, void* d_ws, size_t ws_size,
                              hipStream_t stream)
{
    (void)n_in; (void)d_ws; (void)ws_size; (void)out_size;
    const float* pos = (const float*)d_in[0];
    const int*   Z   = (const int*)d_in[1];
    float*       out = (float*)d_out;

    SoapConsts cc;

    // S overlap matrix, 0-based: S[i][j] = sqrt((7+2i)(7+2j))/(7+i+j)
    double A[8][8], V[8][8];
    for (int i = 0; i < 8; ++i)
        for (int j = 0; j < 8; ++j)
            A[i][j] = sqrt((double)(7 + 2 * i) * (double)(7 + 2 * j)) / (double)(7 + i + j);

    // cyclic Jacobi eigendecomposition (SPD 8x8)
    for (int i = 0; i < 8; ++i)
        for (int j = 0; j < 8; ++j) V[i][j] = (i == j) ? 1.0 : 0.0;
    for (int sweep = 0; sweep < 64; ++sweep) {
        double off = 0.0;
        for (int p = 0; p < 8; ++p)
            for (int q = p + 1; q < 8; ++q) off += A[p][q] * A[p][q];
        if (off < 1e-26) break;
        for (int p = 0; p < 8; ++p) {
            for (int q = p + 1; q < 8; ++q) {
                double apq = A[p][q];
                if (fabs(apq) < 1e-300) continue;
                double theta = (A[q][q] - A[p][p]) / (2.0 * apq);
                double tv = ((theta >= 0.0) ? 1.0 : -1.0) / (fabs(theta) + sqrt(theta * theta + 1.0));
                double c = 1.0 / sqrt(tv * tv + 1.0), s = tv * c;
                for (int k = 0; k < 8; ++k) { double a0 = A[k][p], a1 = A[k][q];
                    A[k][p] = c * a0 - s * a1; A[k][q] = s * a0 + c * a1; }
                for (int k = 0; k < 8; ++k) { double a0 = A[p][k], a1 = A[q][k];
                    A[p][k] = c * a0 - s * a1; A[q][k] = s * a0 + c * a1; }
                for (int k = 0; k < 8; ++k) { double v0 = V[k][p], v1 = V[k][q];
                    V[k][p] = c * v0 - s * v1; V[k][q] = s * v0 + c * v1; }
            }
        }
    }
    double w[8];
    for (int i = 0; i < 8; ++i) w[i] = A[i][i];
    double Wo[8][8];
    for (int i = 0; i < 8; ++i)
        for (int j = 0; j < 8; ++j) {
            double sum = 0.0;
            for (int k = 0; k < 8; ++k) sum += V[i][k] * V[j][k] / sqrt(w[k]);
            Wo[i][j] = sum;
        }
    // fold radial normalization into W: norm[k] = sqrt(5^(2k+7)/(2k+7))
    for (int k = 0; k < 8; ++k) {
        double nk = sqrt(pow(5.0, (double)(2 * k + 7)) / (double)(2 * k + 7));
        for (int n = 0; n < 8; ++n) cc.Wn[n * 8 + k] = (float)(Wo[n][k] / nk);
    }

    const double PI = M_PI, F4 = 4.0 * PI;
    const double yc[25] = {
        0.5 * sqrt(1.0 / PI),
        sqrt(3.0 / (4.0 * PI)), sqrt(3.0 / (4.0 * PI)), sqrt(3.0 / (4.0 * PI)),
        0.5 * sqrt(15.0 / PI), 0.5 * sqrt(15.0 / PI), 0.25 * sqrt(5.0 / PI),
        0.5 * sqrt(15.0 / PI), 0.25 * sqrt(15.0 / PI),
        0.25 * sqrt(35.0 / (2.0 * PI)), 0.5 * sqrt(105.0 / PI), 0.25 * sqrt(21.0 / (2.0 * PI)),
        0.25 * sqrt(7.0 / PI), 0.25 * sqrt(21.0 / (2.0 * PI)), 0.25 * sqrt(105.0 / PI),
        0.25 * sqrt(35.0 / (2.0 * PI)),
        0.75 * sqrt(35.0 / PI), 0.75 * sqrt(35.0 / (2.0 * PI)), 0.75 * sqrt(5.0 / PI),
        0.75 * sqrt(5.0 / (2.0 * PI)), (3.0 / 16.0) * sqrt(1.0 / PI),
        0.75 * sqrt(5.0 / (2.0 * PI)), (3.0 / 8.0) * sqrt(5.0 / PI),
        0.75 * sqrt(35.0 / (2.0 * PI)), (3.0 / 16.0) * sqrt(35.0 / PI)
    };
    for (int m = 0; m < 25; ++m) cc.ycoef[m] = (float)(yc[m] * F4);
    for (int m = 25; m < 32; ++m) cc.ycoef[m] = 0.0f;
    for (int l = 0; l < 5; ++l) cc.psn[l] = (float)(PI * sqrt(8.0 / (double)(2 * l + 1)));
    for (int l = 5; l < 8; ++l) cc.psn[l] = 0.0f;

    const int natoms = in_sizes[1];  // B*N = 2048
    soap_wmma_kernel<<<dim3(natoms), dim3(128), 0, stream>>>(pos, Z, out, cc);
}